// MCHN_34050500723493
// MI455X (gfx1250) — compile-verified
//
#include <hip/hip_runtime.h>
#include <math.h>
#include <stdint.h>

// Problem constants from the reference (cannot read device-side `iterations`
// during graph capture; sizes are fixed by the reference file).
#define BB      16
#define NN      16384
#define DD      512
#define ITERS   10
#define CHUNKS  128                 // workgroups per batch row (split over N)
#define NPW     (NN / CHUNKS)       // 128 patterns per workgroup
#define NTILES  (NPW / 16)          // 8 tiles of 16 patterns
#define TP      (DD + 4)            // padded LDS row stride (floats) -> no bank conflicts
#define PSTRIDE (DD + 2)            // partial record: m, l, acc[DD]

typedef float v2f __attribute__((ext_vector_type(2)));
typedef float v8f __attribute__((ext_vector_type(8)));

// Generic pointers to LDS carry the wave-relative LDS byte offset in the low
// 32 bits (flat-address aperture mapping, ISA 10.2) — exactly what the async
// load's LDS-address VGPR wants.
__device__ __forceinline__ uint32_t lds_off_u32(const void* p) {
  return (uint32_t)(uintptr_t)p;
}

// Stream one 16 x 512 f32 tile (32 KB) of `means` into LDS via the CDNA5
// async-to-LDS path. 256 threads x 8 x b128 = 8 vmem instructions per wave.
__device__ __forceinline__ void async_tile_load(const float* g_base, uint32_t lds_base, int tid) {
#pragma unroll
  for (int i = 0; i < 8; ++i) {
    const int f4  = tid + i * 256;   // float4 index within the 16x512 tile
    const int row = f4 >> 7;         // 128 float4 per row
    const int c4  = f4 & 127;
    const float*  g = g_base + row * DD + c4 * 4;
    const uint32_t l = lds_base + (uint32_t)((row * TP + c4 * 4) * 4);
    asm volatile("global_load_async_to_lds_b128 %0, %1, off"
                 :: "v"(l), "v"(g) : "memory");
  }
}

// Pass 1: per (b, chunk-of-128-patterns) workgroup, one streaming pass over
// means with online softmax. Scores via v_wmma_f32_16x16x4_f32.
__global__ __launch_bounds__(256) void hopfield_pass1(const float* __restrict__ z_src,
                                                      const float* __restrict__ means,
                                                      const float* __restrict__ beta,
                                                      float* __restrict__ part) {
  __shared__ float tile_s[2][16 * TP];   // double-buffered means tile (64.5 KB)
  __shared__ float z_s[DD];
  __shared__ float wsc_s[8][16];         // per-wave partial scores
  __shared__ float sfin_s[16];           // reduced scores for current tile

  const int tid   = threadIdx.x;
  const int b     = blockIdx.x >> 7;            // CHUNKS == 128
  const int chunk = blockIdx.x & (CHUNKS - 1);
  const float* mb = means + ((size_t)b * NN + (size_t)chunk * NPW) * DD;

  // stage z[b,:] into LDS (visible after the first barrier below)
  z_s[tid]       = z_src[b * DD + tid];
  z_s[tid + 256] = z_src[b * DD + tid + 256];
  const float betav = beta[0];

  const int lane = tid & 31;
  const int wave = tid >> 5;
  const int mrow = lane & 15;   // A-matrix row owned by this lane
  const int hi   = lane >> 4;   // 0: K=0..1 slots, 1: K=2..3 slots (ISA A/B layout)

  float m = -INFINITY, l = 0.f, acc_a = 0.f, acc_b = 0.f;

  const uint32_t off0 = lds_off_u32(&tile_s[0][0]);
  const uint32_t off1 = lds_off_u32(&tile_s[1][0]);

  async_tile_load(mb, off0, tid);

  for (int tt = 0; tt < NTILES; ++tt) {
    const int buf = tt & 1;
    if (tt + 1 < NTILES) {
      // prefetch next tile into the other buffer, then retire current tile's
      // 8 in-flight asyncs (they complete in issue order)
      async_tile_load(mb + (size_t)(tt + 1) * 16 * DD, buf ? off0 : off1, tid);
      asm volatile("s_wait_asynccnt 8" ::: "memory");
    } else {
      asm volatile("s_wait_asynccnt 0" ::: "memory");
    }
    __syncthreads();   // tile[buf] (and z_s on tt==0) visible to all waves

    const float* tb = buf ? &tile_s[1][0] : &tile_s[0][0];

    // ---- 16 scores = means_tile(16xD) . z via V_WMMA_F32_16X16X4_F32 ----
    // wave w covers K in [64w, 64w+64): 16 chained WMMAs; B's 16 columns all
    // broadcast z[k], so every column of D holds the same partial dot.
    {
      v8f c = {0.f, 0.f, 0.f, 0.f, 0.f, 0.f, 0.f, 0.f};
      const float* tp = tb + mrow * TP;
      const int kb = wave * 64;
#pragma unroll
      for (int kk = 0; kk < 16; ++kk) {
        const int k = kb + kk * 4 + 2 * hi;
        v2f a;  a.x  = tp[k];    a.y  = tp[k + 1];    // A[mrow][k..k+1 | k+2..k+3]
        v2f bz; bz.x = z_s[k];   bz.y = z_s[k + 1];   // B rows broadcast z
        c = __builtin_amdgcn_wmma_f32_16x16x4_f32(false, a, false, bz,
                                                  (short)0, c, false, false);
      }
      if (mrow == 0) {           // lane 0 -> rows 0..7, lane 16 -> rows 8..15
#pragma unroll
        for (int r = 0; r < 8; ++r) wsc_s[wave][hi * 8 + r] = c[r];
      }
    }
    __syncthreads();

    if (tid < 16) {              // reduce 8 per-wave K-slices
      float s = 0.f;
#pragma unroll
      for (int w = 0; w < 8; ++w) s += wsc_s[w][tid];
      sfin_s[tid] = betav * s;
    }
    __syncthreads();

    // ---- online-softmax update; thread owns d = tid and d = tid + 256 ----
    float s[16], tmax = -INFINITY;
#pragma unroll
    for (int j = 0; j < 16; ++j) { s[j] = sfin_s[j]; tmax = fmaxf(tmax, s[j]); }
    const float nm   = fmaxf(m, tmax);
    const float corr = __expf(m - nm);   // exp(-inf)=0 handles the first tile
    float aa = acc_a * corr, ab = acc_b * corr, ps = 0.f;
#pragma unroll
    for (int j = 0; j < 16; ++j) {
      const float p = __expf(s[j] - nm);
      ps += p;
      aa = fmaf(p, tb[j * TP + tid], aa);         // conflict-free: d == lane
      ab = fmaf(p, tb[j * TP + tid + 256], ab);
    }
    l = l * corr + ps;
    acc_a = aa; acc_b = ab; m = nm;
    __syncthreads();   // everyone done with tile[buf] before it is refilled
  }

  float* op = part + (size_t)blockIdx.x * PSTRIDE;
  if (tid == 0) { op[0] = m; op[1] = l; }
  op[2 + tid]       = acc_a;
  op[2 + 256 + tid] = acc_b;
}

// Pass 2: merge the CHUNKS online-softmax partials per batch row (no re-read
// of means): z[d] = sum_i e^{m_i-M} acc_i[d] / sum_i e^{m_i-M} l_i
__global__ __launch_bounds__(256) void hopfield_merge(const float* __restrict__ part,
                                                      float* __restrict__ z_dst) {
  __shared__ float sm[CHUNKS], sl[CHUNKS];
  const int b = blockIdx.x, tid = threadIdx.x;
  const float* pb = part + (size_t)b * CHUNKS * PSTRIDE;
  if (tid < CHUNKS) {
    sm[tid] = pb[(size_t)tid * PSTRIDE];
    sl[tid] = pb[(size_t)tid * PSTRIDE + 1];
  }
  __syncthreads();
  float M = -INFINITY;
  for (int i = 0; i < CHUNKS; ++i) M = fmaxf(M, sm[i]);
  float L = 0.f;
  for (int i = 0; i < CHUNKS; ++i) L += sl[i] * __expf(sm[i] - M);
  float za = 0.f, zb = 0.f;
  for (int i = 0; i < CHUNKS; ++i) {
    const float w = __expf(sm[i] - M);
    const float* a = pb + (size_t)i * PSTRIDE + 2;
    za = fmaf(w, a[tid], za);
    zb = fmaf(w, a[tid + 256], zb);
  }
  const float inv = 1.f / L;
  z_dst[b * DD + tid]       = za * inv;
  z_dst[b * DD + tid + 256] = zb * inv;
}

extern "C" void kernel_launch(void* const* d_in, const int* in_sizes, int n_in,
                              void* d_out, int out_size, void* d_ws, size_t ws_size,
                              hipStream_t stream) {
  (void)in_sizes; (void)n_in; (void)out_size; (void)ws_size;
  // inputs: x (unused), z, means, beta, iterations (fixed at 10)
  const float* z0    = (const float*)d_in[1];
  const float* means = (const float*)d_in[2];
  const float* beta  = (const float*)d_in[3];

  float* ws   = (float*)d_ws;
  float* zb0  = ws;                    // B*D
  float* zb1  = ws + BB * DD;          // B*D
  float* part = ws + 2 * BB * DD;      // B*CHUNKS*PSTRIDE (~4.2 MB)

  for (int it = 0; it < ITERS; ++it) {
    const float* zsrc = (it == 0) ? z0 : ((it & 1) ? zb0 : zb1);
    float* zdst = (it == ITERS - 1) ? (float*)d_out : ((it & 1) ? zb1 : zb0);
    hopfield_pass1<<<dim3(BB * CHUNKS), dim3(256), 0, stream>>>(zsrc, means, beta, part);
    hopfield_merge<<<dim3(BB), dim3(256), 0, stream>>>(part, zdst);
  }
}